// EdgeCrossingsLoss_84945863180900
// MI455X (gfx1250) — compile-verified
//
#include <hip/hip_runtime.h>
#include <hip/hip_bf16.h>

typedef __attribute__((ext_vector_type(2))) float v2f;
typedef __attribute__((ext_vector_type(8))) float v8f;

#define KNN 16
#define EPSF 1e-5f

// ---------------------------------------------------------------------------
// Kernel 1: per-face barycenter + squared norm; zero the output scalar.
// ---------------------------------------------------------------------------
__global__ void k_bary(const float* __restrict__ vertices,
                       const int* __restrict__ faces,
                       float* __restrict__ bary,
                       float* __restrict__ sq,
                       float* __restrict__ out,
                       int F)
{
    int f = blockIdx.x * blockDim.x + threadIdx.x;
    if (f == 0) out[0] = 0.0f;
    if (f >= F) return;
    int i0 = faces[f * 3 + 0];
    int i1 = faces[f * 3 + 1];
    int i2 = faces[f * 3 + 2];
    const float third = 1.0f / 3.0f;
    float bx = (vertices[i0 * 3 + 0] + vertices[i1 * 3 + 0] + vertices[i2 * 3 + 0]) * third;
    float by = (vertices[i0 * 3 + 1] + vertices[i1 * 3 + 1] + vertices[i2 * 3 + 1]) * third;
    float bz = (vertices[i0 * 3 + 2] + vertices[i1 * 3 + 2] + vertices[i2 * 3 + 2]) * third;
    bary[f * 3 + 0] = bx;
    bary[f * 3 + 1] = by;
    bary[f * 3 + 2] = bz;
    sq[f] = bx * bx + by * by + bz * bz;
}

// ---------------------------------------------------------------------------
// Kernel 2: one wave32 per 16-row tile.
//   - v_wmma_f32_16x16x4_f32 computes a 16x16 Gram tile (K=3 padded to 4).
//   - d2 tile staged through LDS once per tile (wide b128 reads back).
//   - Lane (m, half) keeps a REGISTER-resident top-16 list over columns
//     [half*8, half*8+8) of each tile; the two partial lists per row are a
//     superset of the global top-16 and get merged once at the end.
//
// 32-bit 16x4 A layout (ISA 7.12.2): lanes 0-15 hold {K=0,K=1} for M=lane,
// lanes 16-31 hold {K=2,K=3} for M=lane-16. B (4x16) mirrors it.
// C/D: VGPR r -> (M=r, N=lane) for lanes 0-15, (M=r+8, N=lane-16) for 16-31.
// ---------------------------------------------------------------------------
__global__ __launch_bounds__(32) void k_topk(const float* __restrict__ bary,
                                             const float* __restrict__ sq,
                                             int* __restrict__ nbr,
                                             int F)
{
    __shared__ float tile[16 * 16];
    __shared__ float mv[16 * 32];   // merged candidate values per row
    __shared__ int   mi[16 * 32];   // merged candidate indices per row

    const int lane = threadIdx.x;
    const int hf   = lane >> 4;    // 0: lanes 0-15, 1: lanes 16-31
    const int mr   = lane & 15;    // row owned by this lane (both halves)
    const int I    = blockIdx.x * 16;
    const float INF = __builtin_inff();

    // A tile (rows I..I+15), loaded once.
    v2f a;
    {
        int row = I + mr;
        float b0 = 0.f, b1 = 0.f, b2 = 0.f;
        if (row < F) { b0 = bary[row * 3 + 0]; b1 = bary[row * 3 + 1]; b2 = bary[row * 3 + 2]; }
        a.x = hf ? b2 : b0;        // K=0 (lo half) / K=2 (hi half)
        a.y = hf ? 0.0f : b1;      // K=1 (lo half) / K=3 pad (hi half)
    }
    float sqI[8];
#pragma unroll
    for (int r = 0; r < 8; ++r) {
        int rr = I + r + hf * 8;
        sqI[r] = (rr < F) ? sq[rr] : 0.0f;
    }

    // Register-resident partial top-16 list for this (row, column-half).
    float tv[KNN];
    int   ti[KNN];
#pragma unroll
    for (int t = 0; t < KNN; ++t) { tv[t] = INF; ti[t] = 0x7fffffff; }
    float maxVal = INF;
    int   maxPos = 0;

    const int nT = (F + 15) >> 4;
    for (int Jt = 0; Jt < nT; ++Jt) {
        const int J   = Jt * 16;
        const int col = J + mr;

        // Prefetch next column tile's barycenters (L2-resident; hides WGP$ miss).
        if (Jt + 1 < nT) __builtin_prefetch(&bary[(size_t)(J + 16) * 3 + mr], 0, 3);

        v2f b;
        float c0 = 0.f, c1 = 0.f, c2 = 0.f, sqJ = 0.f;
        if (col < F) { c0 = bary[col * 3 + 0]; c1 = bary[col * 3 + 1]; c2 = bary[col * 3 + 2]; sqJ = sq[col]; }
        b.x = hf ? c2 : c0;
        b.y = hf ? 0.0f : c1;

        v8f acc = {};
        acc = __builtin_amdgcn_wmma_f32_16x16x4_f32(
            /*neg_a=*/false, a, /*neg_b=*/false, b,
            /*c_mod=*/(short)0, acc, /*reuse_a=*/false, /*reuse_b=*/false);

#pragma unroll
        for (int r = 0; r < 8; ++r) {
            int m = r + hf * 8;
            float d2 = (col < F) ? (sqI[r] + sqJ - 2.0f * acc[r]) : INF;
            tile[m * 16 + mr] = d2;
        }
        __syncthreads();

        // Pull this lane's 8 candidates back with wide LDS reads (b128 x2).
        float rowv[8];
#pragma unroll
        for (int n = 0; n < 8; ++n) rowv[n] = tile[mr * 16 + hf * 8 + n];
        __syncthreads();   // tile free for next iteration; inserts below are reg-only

#pragma unroll
        for (int n = 0; n < 8; ++n) {
            float cand = rowv[n];
            if (cand < maxVal) {
                int idx = J + hf * 8 + n;
#pragma unroll
                for (int t = 0; t < KNN; ++t) {
                    if (t == maxPos) { tv[t] = cand; ti[t] = idx; }
                }
                float nv = tv[0]; int np = 0;
#pragma unroll
                for (int t = 1; t < KNN; ++t) {
                    if (tv[t] > nv) { nv = tv[t]; np = t; }
                }
                maxVal = nv; maxPos = np;
            }
        }
    }

    // Dump both partial lists per row to LDS and merge 32 -> 16.
    {
        int base = mr * 32 + hf * KNN;
#pragma unroll
        for (int t = 0; t < KNN; ++t) { mv[base + t] = tv[t]; mi[base + t] = ti[t]; }
    }
    __syncthreads();

    if (lane < 16) {
        int row = I + lane;
        if (row < F) {
            unsigned used = 0u;
            for (int t = 0; t < KNN; ++t) {
                float bv = INF; int bi = 0x7fffffff; int bp = 0;
                for (int u = 0; u < 32; ++u) {
                    if (used & (1u << u)) continue;
                    float v  = mv[lane * 32 + u];
                    int   ix = mi[lane * 32 + u];
                    // (value, index) lexicographic: matches top_k tie behavior
                    if (v < bv || (v == bv && ix < bi)) { bv = v; bi = ix; bp = u; }
                }
                used |= 1u << bp;
                nbr[row * KNN + t] = bi;
            }
        }
    }
}

// ---------------------------------------------------------------------------
// Kernel 3: per (face, neighbor-slot) edge-pair crossing tests + reduction.
// ---------------------------------------------------------------------------
__device__ __forceinline__ void load_edges(const float* __restrict__ v,
                                           const int* __restrict__ faces,
                                           int f, float s[3][3], float d[3][3])
{
    float p[3][3];
#pragma unroll
    for (int k = 0; k < 3; ++k) {
        int vi = faces[f * 3 + k];
        p[k][0] = v[vi * 3 + 0];
        p[k][1] = v[vi * 3 + 1];
        p[k][2] = v[vi * 3 + 2];
    }
    // starts = pos[:, [0,0,1]], dirs = pos[:, [1,2,2]] - starts
#pragma unroll
    for (int c = 0; c < 3; ++c) {
        s[0][c] = p[0][c]; d[0][c] = p[1][c] - p[0][c];
        s[1][c] = p[0][c]; d[1][c] = p[2][c] - p[0][c];
        s[2][c] = p[1][c]; d[2][c] = p[2][c] - p[1][c];
    }
}

__global__ __launch_bounds__(256) void k_cross(const float* __restrict__ vertices,
                                               const int* __restrict__ faces,
                                               const float* __restrict__ prob,
                                               const int* __restrict__ nbr,
                                               float* __restrict__ out,
                                               int F)
{
    __shared__ float red[256];
    int gid = blockIdx.x * blockDim.x + threadIdx.x;
    float w = 0.0f;

    if (gid < F * KNN) {
        int i = gid / KNN;
        int j = nbr[gid];
        if (j >= 0 && j < F && j != i) {
            float s1[3][3], d1[3][3], s2[3][3], d2[3][3];
            load_edges(vertices, faces, i, s1, d1);
            load_edges(vertices, faces, j, s2, d2);
            int hits = 0;
#pragma unroll
            for (int e1 = 0; e1 < 3; ++e1) {
#pragma unroll
                for (int e2 = 0; e2 < 3; ++e2) {
                    // cr = cross(dir_i, dir_j)
                    float cx = d1[e1][1] * d2[e2][2] - d1[e1][2] * d2[e2][1];
                    float cy = d1[e1][2] * d2[e2][0] - d1[e1][0] * d2[e2][2];
                    float cz = d1[e1][0] * d2[e2][1] - d1[e1][1] * d2[e2][0];
                    float fx = s2[e2][0] - s1[e1][0];
                    float fy = s2[e2][1] - s1[e1][1];
                    float fz = s2[e2][2] - s1[e1][2];
                    float num = fabsf(fx * cx + fy * cy + fz * cz);
                    float den = sqrtf(cx * cx + cy * cy + cz * cz);
                    float dist = num / den;  // 0/0 -> NaN -> compare false (matches jnp)
                    float gx = d2[e2][0] - d1[e1][0];
                    float gy = d2[e2][1] - d1[e1][1];
                    float gz = d2[e2][2] - d1[e1][2];
                    float same = fx * fx + fy * fy + fz * fz + gx * gx + gy * gy + gz * gz;
                    bool hit = (dist < EPSF) && !(same < EPSF * EPSF);
                    hits += hit ? 1 : 0;
                }
            }
            w = prob[i] * (float)hits;
        }
    }

    red[threadIdx.x] = w;
    __syncthreads();
    for (int s = blockDim.x >> 1; s > 0; s >>= 1) {
        if ((int)threadIdx.x < s) red[threadIdx.x] += red[threadIdx.x + s];
        __syncthreads();
    }
    if (threadIdx.x == 0) atomicAdd(out, red[0] / (float)F);
}

// ---------------------------------------------------------------------------
extern "C" void kernel_launch(void* const* d_in, const int* in_sizes, int n_in,
                              void* d_out, int out_size, void* d_ws, size_t ws_size,
                              hipStream_t stream)
{
    const float* vertices = (const float*)d_in[0];
    const int*   faces    = (const int*)d_in[1];
    const float* prob     = (const float*)d_in[2];
    float*       out      = (float*)d_out;

    const int F = in_sizes[1] / 3;

    // Workspace layout: bary (F*3 f32) | sq (F f32) | nbr (F*KNN i32)  ~= 800 KB
    float* bary = (float*)d_ws;
    float* sq   = bary + (size_t)F * 3;
    int*   nbr  = (int*)(sq + F);

    {
        int t = 256, g = (F + t - 1) / t;
        k_bary<<<g, t, 0, stream>>>(vertices, faces, bary, sq, out, F);
    }
    {
        int rowTiles = (F + 15) / 16;
        k_topk<<<rowTiles, 32, 0, stream>>>(bary, sq, nbr, F);
    }
    {
        int pairs = F * KNN;
        int t = 256, g = (pairs + t - 1) / t;
        k_cross<<<g, t, 0, stream>>>(vertices, faces, prob, nbr, out, F);
    }
}